// GPTNeoXLayerTalking_74998718923387
// MI455X (gfx1250) — compile-verified
//
#include <hip/hip_runtime.h>
#include <math.h>

// Problem constants (B=1, G=1)
#define T_DIM 2048
#define S_DIM 2048
#define D_DIM 2048
#define NH    16
#define WS_ROW 320      // floats per token of dynamic weights
#define XPITCH 2052     // LDS row pitch (floats): 16B-aligned rows for async B128,
                        // and pitch%64==4 -> D-layout writeback halves 32 banks apart

typedef __attribute__((ext_vector_type(8))) float v8f;
typedef __attribute__((ext_vector_type(2))) float v2f;

// Workspace layout per token t (floats, stride WS_ROW):
//   0: pre_qw1[2][16]   32: pre_qw2[2][16]   64: pre_kw1[2][16]   96: pre_kw2[2][16]
// 128: pre_qdd[16]     144: pre_kdd[16]
// 160: post_qw1        192: post_qw2        224: post_kw1        256: post_kw2
// 288: post_qdd        304: post_kdd

// ---------------- Kernel 1: per-token dynamic weights ----------------
__global__ __launch_bounds__(256)
void dynw_kernel(const float* __restrict__ q, const float* __restrict__ dw1,
                 const float* __restrict__ qkw, const float* __restrict__ dd,
                 float* __restrict__ ws) {
  __shared__ float qs[D_DIM];
  __shared__ float dwh[256];
  __shared__ float wl[256];
  __shared__ float rms[8];
  const int t = blockIdx.x, tid = threadIdx.x;
  const float* qrow = q + (size_t)t * D_DIM;
  for (int d = tid; d < D_DIM; d += 256) qs[d] = qrow[d];
  __syncthreads();

  // dw_hidden[c][k] = gelu(q[t,:] . dw1[:, c, k]);  tid = c*64+k (coalesced over tid)
  float acc = 0.f;
  for (int d = 0; d < D_DIM; ++d) acc = fmaf(qs[d], dw1[(size_t)d * 256 + tid], acc);
  dwh[tid] = 0.5f * acc * (1.f + erff(acc * 0.70710678118654752f)); // exact gelu

  // ddv[j] = tanh(q[t,:] . dd[:, j]), j = c2*16+mm -> {pre_qdd, pre_kdd, post_qdd, post_kdd}
  if (tid < 64) {
    float a2 = 0.f;
    for (int d = 0; d < D_DIM; ++d) a2 = fmaf(qs[d], dd[(size_t)d * 64 + tid], a2);
    const int ddoff[4] = {128, 144, 288, 304};
    ws[(size_t)t * WS_ROW + ddoff[tid >> 4] + (tid & 15)] = tanhf(a2);
  }
  __syncthreads();

  // w[c][i2][m] = dwh[c][:] . qkw[c,:,i2,m];  tid = c*64 + i2*16 + m
  const int c = tid >> 6, rem = tid & 63;
  float a3 = 0.f;
  for (int k = 0; k < 64; ++k)
    a3 = fmaf(dwh[c * 64 + k], qkw[(size_t)(c * 64 + k) * 64 + rem], a3);
  wl[tid] = a3;
  __syncthreads();

  // rmsnorm_noscale over M for w1 (i2 < 2)
  if (tid < 8) {
    const int cc = tid >> 1, ii = tid & 1;
    float s2 = 0.f;
    for (int m = 0; m < 16; ++m) { float v = wl[cc * 64 + ii * 16 + m]; s2 += v * v; }
    rms[tid] = rsqrtf(s2 * (1.f / 16.f) + 1e-6f);
  }
  __syncthreads();

  const int i2 = rem >> 4, m = rem & 15;
  const int off1[4] = {0, 64, 160, 224};   // qw1/kw1 (pre, then post) per c
  const int off2[4] = {32, 96, 192, 256};  // qw2/kw2
  float val = wl[tid];
  const size_t base = (size_t)t * WS_ROW;
  if (i2 < 2) ws[base + off1[c] + i2 * 16 + m] = val * rms[c * 2 + i2];
  else        ws[base + off2[c] + (i2 - 2) * 16 + m] = val;
}

// ------- Kernel 2: fused pre-mix -> causal softmax -> post-mix, one WG per row t -------
__global__ __launch_bounds__(512)
void fused_row_kernel(const float* __restrict__ logits, const float* __restrict__ w_pre,
                      const float* __restrict__ w_post, const float* __restrict__ ws,
                      float* __restrict__ out) {
  __shared__ float xbuf[NH * XPITCH];     // 16 x 2048 row of all heads (pitch 2052) ~128 KB
  __shared__ float weff_pre[256];
  __shared__ float weff_post[256];
  const int t = blockIdx.x, tid = threadIdx.x;
  const float* wst = ws + (size_t)t * WS_ROW;

  // Stage logits row t of all 16 heads into LDS via CDNA5 async VMEM->LDS path.
  // Per-lane 16B transfers, data never touches VGPRs; tracked on ASYNCcnt.
  for (int it = 0; it < 16; ++it) {
    const int idx4 = tid + it * 512;                // 0..8191 float4 tiles
    const int n = idx4 >> 9, sq = idx4 & 511;       // head, float4-column
    const float* g = logits + ((size_t)n * T_DIM + t) * S_DIM + (size_t)sq * 4;
    const unsigned l = (unsigned)(uintptr_t)(xbuf + n * XPITCH + sq * 4);
    asm volatile("global_load_async_to_lds_b128 %0, %1, off"
                 :: "v"(l), "v"(g) : "memory");
  }

  // Build Weff[m][n] = (w + I)[m][n] + sum_i qw1[i][m]*qw2[i][n] + (m==n)*qdd[n]
  {
    const int idx = tid & 255, m = idx >> 4, n = idx & 15;
    const bool post = tid >= 256;
    const float* w0  = post ? w_post : w_pre;
    const float* qw1 = wst + (post ? 160 : 0);
    const float* qw2 = wst + (post ? 192 : 32);
    const float* qdd = wst + (post ? 288 : 128);
    float v = w0[idx] + qw1[m] * qw2[n] + qw1[16 + m] * qw2[16 + n];
    if (m == n) v += 1.f + qdd[n];
    (post ? weff_post : weff_pre)[idx] = v;
  }

  asm volatile("s_wait_asynccnt 0" ::: "memory");   // this wave's async fills done
  __syncthreads();                                  // all waves' fills visible

  const int lane = tid & 31, wv = tid >> 5;
  const int half = lane >> 4, col = lane & 15;

  // A operands (Weff^T slices) for the 4 chained K=4 WMMAs; loaded once per wave.
  // A[row=n(lane&15), k=v+2*half] of slice kk  ->  Weff[kk*4 + v + 2*half][n]
  v2f a_pre[4], a_post[4];
  for (int kk = 0; kk < 4; ++kk)
    for (int v = 0; v < 2; ++v) {
      a_pre[kk][v]  = weff_pre [(kk * 4 + v + 2 * half) * 16 + col];
      a_post[kk][v] = weff_post[(kk * 4 + v + 2 * half) * 16 + col];
    }

  // ---------- pre cross-head projection (in place) ----------
  for (int ch = wv; ch < S_DIM / 16; ch += 16) {
    const int s = ch * 16 + col;
    const float* wr = ws + (size_t)s * WS_ROW;   // per-s k-term weights (L2-resident)
    float xc[16];
    for (int m = 0; m < 16; ++m) xc[m] = xbuf[m * XPITCH + s];
    float h0 = 0.f, h1 = 0.f;                    // h[i] = sum_m kw1[s,i,m]*x[m,s]
    for (int m = 0; m < 16; ++m) {
      h0 = fmaf(wr[64 + m], xc[m], h0);
      h1 = fmaf(wr[80 + m], xc[m], h1);
    }
    v8f c;                                       // C = k-terms in WMMA D layout
    for (int r = 0; r < 8; ++r) {
      const int n = r + 8 * half;
      c[r] = wr[96 + n] * h0 + wr[112 + n] * h1 + xc[n] * wr[144 + n];
    }
    for (int kk = 0; kk < 4; ++kk) {             // D = Weff^T * X + C, K=16 via 4x K=4
      v2f b; b[0] = xc[kk * 4 + 2 * half]; b[1] = xc[kk * 4 + 1 + 2 * half];
      c = __builtin_amdgcn_wmma_f32_16x16x4_f32(false, a_pre[kk], false, b,
                                                (short)0, c, false, false);
    }
    for (int r = 0; r < 8; ++r) xbuf[(r + 8 * half) * XPITCH + s] = c[r];
  }
  __syncthreads();

  // ---------- causal softmax over S, one wave32 per head ----------
  {
    float* row = xbuf + wv * XPITCH;
    float mx = -3.0e38f;
    for (int s = lane; s <= t; s += 32) mx = fmaxf(mx, row[s]);
    for (int o = 16; o; o >>= 1) mx = fmaxf(mx, __shfl_xor(mx, o, 32));
    float sum = 0.f;
    for (int s = lane; s <= t; s += 32) { float e = expf(row[s] - mx); row[s] = e; sum += e; }
    for (int o = 16; o; o >>= 1) sum += __shfl_xor(sum, o, 32);
    const float inv = 1.f / sum;
    for (int s = lane; s < S_DIM; s += 32) row[s] = (s <= t) ? row[s] * inv : 0.f;
  }
  __syncthreads();

  // ---------- post cross-head projection, stream to HBM (non-temporal) ----------
  for (int ch = wv; ch < S_DIM / 16; ch += 16) {
    const int s = ch * 16 + col;
    const float* wr = ws + (size_t)s * WS_ROW;
    float xc[16];
    for (int m = 0; m < 16; ++m) xc[m] = xbuf[m * XPITCH + s];
    float h0 = 0.f, h1 = 0.f;
    for (int m = 0; m < 16; ++m) {
      h0 = fmaf(wr[224 + m], xc[m], h0);
      h1 = fmaf(wr[240 + m], xc[m], h1);
    }
    v8f c;
    for (int r = 0; r < 8; ++r) {
      const int n = r + 8 * half;
      c[r] = wr[256 + n] * h0 + wr[272 + n] * h1 + xc[n] * wr[304 + n];
    }
    for (int kk = 0; kk < 4; ++kk) {
      v2f b; b[0] = xc[kk * 4 + 2 * half]; b[1] = xc[kk * 4 + 1 + 2 * half];
      c = __builtin_amdgcn_wmma_f32_16x16x4_f32(false, a_post[kk], false, b,
                                                (short)0, c, false, false);
    }
    for (int r = 0; r < 8; ++r) {
      const int n = r + 8 * half;
      __builtin_nontemporal_store(c[r], &out[((size_t)n * T_DIM + t) * S_DIM + s]);
    }
  }
}

extern "C" void kernel_launch(void* const* d_in, const int* in_sizes, int n_in,
                              void* d_out, int out_size, void* d_ws, size_t ws_size,
                              hipStream_t stream) {
  const float* query_vec = (const float*)d_in[0]; // (1, 2048, 2048)
  const float* logits    = (const float*)d_in[1]; // (1, 16, 2048, 2048)
  const float* dw1       = (const float*)d_in[2]; // (2048, 1, 4, 64)
  const float* qkw       = (const float*)d_in[3]; // (1, 4, 64, 4, 16)
  const float* dd        = (const float*)d_in[4]; // (2048, 1, 64)
  const float* w_pre     = (const float*)d_in[5]; // (1, 16, 16)
  const float* w_post    = (const float*)d_in[6]; // (1, 16, 16)
  float* out = (float*)d_out;
  float* ws  = (float*)d_ws;                      // needs T_DIM*WS_ROW*4 = 2.62 MB

  dynw_kernel<<<T_DIM, 256, 0, stream>>>(query_vec, dw1, qkw, dd, ws);
  fused_row_kernel<<<T_DIM, 512, 0, stream>>>(logits, w_pre, w_post, ws, out);
}